// SplitGNN_5_9783935500858
// MI455X (gfx1250) — compile-verified
//
#include <hip/hip_runtime.h>
#include <hip/hip_bf16.h>

#define B_    32
#define HIST_ 8
#define PRED_ 24
#define N_    256
#define F_    15
#define HID_  64
#define EMH_  32
#define E_    4096
#define D_    17          // F + 2 (pm_last + feat + cn)
#define TT_   32          // HIST + PRED
#define M_    (B_ * N_)   // 8192 rows
#define PRED_SZ (B_ * PRED_ * N_)

typedef float v2f __attribute__((ext_vector_type(2)));
typedef float v8f __attribute__((ext_vector_type(8)));

// ---------------------------------------------------------------------------
// One-time: normalize edge_attr (mean / std with ddof=1) -> ean[E][2]
// ---------------------------------------------------------------------------
__global__ __launch_bounds__(256) void ean_kernel(const float* __restrict__ edge_attr,
                                                  float* __restrict__ ean) {
  __shared__ float red0[256], red1[256];
  __shared__ float mean0, mean1, istd0, istd1;
  int tid = threadIdx.x;
  float s0 = 0.f, s1 = 0.f;
  for (int e = tid; e < E_; e += 256) { s0 += edge_attr[e * 2]; s1 += edge_attr[e * 2 + 1]; }
  red0[tid] = s0; red1[tid] = s1; __syncthreads();
  for (int o = 128; o; o >>= 1) {
    if (tid < o) { red0[tid] += red0[tid + o]; red1[tid] += red1[tid + o]; }
    __syncthreads();
  }
  if (tid == 0) { mean0 = red0[0] / (float)E_; mean1 = red1[0] / (float)E_; }
  __syncthreads();
  s0 = 0.f; s1 = 0.f;
  for (int e = tid; e < E_; e += 256) {
    float d0 = edge_attr[e * 2] - mean0, d1 = edge_attr[e * 2 + 1] - mean1;
    s0 += d0 * d0; s1 += d1 * d1;
  }
  red0[tid] = s0; red1[tid] = s1; __syncthreads();
  for (int o = 128; o; o >>= 1) {
    if (tid < o) { red0[tid] += red0[tid + o]; red1[tid] += red1[tid + o]; }
    __syncthreads();
  }
  if (tid == 0) {
    istd0 = rsqrtf(red0[0] / (float)(E_ - 1));
    istd1 = rsqrtf(red1[0] / (float)(E_ - 1));
  }
  __syncthreads();
  for (int e = tid; e < E_; e += 256) {
    ean[e * 2]     = (edge_attr[e * 2]     - mean0) * istd0;
    ean[e * 2 + 1] = (edge_attr[e * 2 + 1] - mean1) * istd1;
  }
}

// ---------------------------------------------------------------------------
// One-time: CSR of edges sorted by target row (tgt = edge_index[1][e])
// ---------------------------------------------------------------------------
__global__ __launch_bounds__(256) void csr_kernel(const int* __restrict__ edge_index,
                                                  int* __restrict__ row_ptr,
                                                  int* __restrict__ csr_col,
                                                  int* __restrict__ csr_eid) {
  __shared__ int cnt[N_];
  __shared__ int offs[N_];
  int tid = threadIdx.x;
  cnt[tid] = 0;
  __syncthreads();
  for (int e = tid; e < E_; e += 256) atomicAdd(&cnt[edge_index[E_ + e]], 1);
  __syncthreads();
  if (tid == 0) {
    int acc = 0;
    for (int i = 0; i < N_; ++i) { row_ptr[i] = acc; offs[i] = acc; acc += cnt[i]; }
    row_ptr[N_] = acc;
  }
  __syncthreads();
  for (int e = tid; e < E_; e += 256) {
    int tg = edge_index[E_ + e];
    int pos = atomicAdd(&offs[tg], 1);
    csr_col[pos] = edge_index[e];  // src
    csr_eid[pos] = e;
  }
}

// ---------------------------------------------------------------------------
// Per step: edge MLP -> e_rep[b][e]
// ---------------------------------------------------------------------------
__global__ __launch_bounds__(256) void edge_kernel(const float* __restrict__ feature,
                                                   const int* __restrict__ edge_index,
                                                   const float* __restrict__ edge_attr,
                                                   const float* __restrict__ ean,
                                                   const float* __restrict__ wind_mean,
                                                   const float* __restrict__ wind_std,
                                                   const float* __restrict__ W_e1,
                                                   const float* __restrict__ b_e1,
                                                   const float* __restrict__ W_e2,
                                                   const float* __restrict__ b_e2,
                                                   float* __restrict__ e_rep, int t) {
  int id = blockIdx.x * 256 + threadIdx.x;  // B*E total
  int b = id >> 12;         // / E_
  int e = id & (E_ - 1);
  int src = edge_index[e];  // row 0 of (2,E)
  const float* fp = feature + ((size_t)(b * TT_ + (HIST_ + t)) * N_ + src) * F_;
  float sw0 = fp[13] * wind_std[0] + wind_mean[0];
  float sw1 = fp[14] * wind_std[1] + wind_mean[1];
  float dist = edge_attr[e * 2], direc = edge_attr[e * 2 + 1];
  float theta = fabsf(direc - sw1);
  float ew = fmaxf(0.f, 3.f * sw0 * cosf(theta) / dist);
  float a0 = ean[e * 2], a1 = ean[e * 2 + 1];
  float out = b_e2[0];
#pragma unroll
  for (int j = 0; j < EMH_; ++j) {
    float h = a0 * W_e1[j] + a1 * W_e1[EMH_ + j] + ew * W_e1[2 * EMH_ + j] + b_e1[j];
    out += fmaxf(h, 0.f) * W_e2[j];
  }
  e_rep[id] = out;
}

// ---------------------------------------------------------------------------
// Per step: GRU cell with fp32 WMMA (16x16x4).
// Block = 128 threads (4 waves) handles 16 rows; wave g owns gate col-tile g.
// ---------------------------------------------------------------------------
__global__ __launch_bounds__(128) void gru_kernel(const float* __restrict__ pm_hist,
                                                  const float* __restrict__ feature,
                                                  const float* __restrict__ W_ih,
                                                  const float* __restrict__ b_ih,
                                                  const float* __restrict__ W_hh,
                                                  const float* __restrict__ b_hh,
                                                  const float* __restrict__ W_node,
                                                  const float* __restrict__ b_node,
                                                  float* __restrict__ hn,
                                                  const float* __restrict__ cn,
                                                  float* __restrict__ node_out, int t) {
  __shared__ float lds_x[16][20];   // node_in, K padded 17 -> 20
  __shared__ float lds_h[16][HID_]; // previous hidden
  __shared__ float lds_o[16][HID_]; // new hidden (for W_node row-dot)

  const int tid = threadIdx.x;
  const int lane = tid & 31;
  const int g = tid >> 5;            // wave id = gate col tile 0..3
  const int rowbase = blockIdx.x * 16;

  for (int i = tid; i < 16 * 20; i += 128) {
    int r = i / 20, c = i % 20;
    int row = rowbase + r;
    int b = row >> 8, n = row & (N_ - 1);
    float v = 0.f;
    if (c == 0)       v = pm_hist[(b * HIST_ + (HIST_ - 1)) * N_ + n];
    else if (c <= 15) v = feature[((size_t)(b * TT_ + (HIST_ + t)) * N_ + n) * F_ + (c - 1)];
    else if (c == 16) v = cn[row];
    lds_x[r][c] = v;
  }
  for (int i = tid; i < 16 * HID_; i += 128) {
    int r = i >> 6, c = i & (HID_ - 1);
    lds_h[r][c] = hn[(rowbase + r) * HID_ + c];
  }
  __syncthreads();

  const int nlo = lane & 15;              // M for A-frag, N for B-frag
  const int khalf = (lane >> 4) * 2;      // K offset within 4-chunk
  const int cr = g * 16, cz = 64 + g * 16, cnn = 128 + g * 16;

  v8f a_ir = {}, a_iz = {}, a_in = {}, a_hr = {}, a_hz = {}, a_hn = {};

  // gi = node_in @ W_ih^T   (K = 17, padded to 20)
#pragma unroll
  for (int kb = 0; kb < 20; kb += 4) {
    int k0 = kb + khalf, k1 = k0 + 1;
    v2f a; a.x = lds_x[nlo][k0]; a.y = lds_x[nlo][k1];
    v2f br, bz, bn;
    br.x = (k0 < D_) ? W_ih[(cr  + nlo) * D_ + k0] : 0.f;
    br.y = (k1 < D_) ? W_ih[(cr  + nlo) * D_ + k1] : 0.f;
    bz.x = (k0 < D_) ? W_ih[(cz  + nlo) * D_ + k0] : 0.f;
    bz.y = (k1 < D_) ? W_ih[(cz  + nlo) * D_ + k1] : 0.f;
    bn.x = (k0 < D_) ? W_ih[(cnn + nlo) * D_ + k0] : 0.f;
    bn.y = (k1 < D_) ? W_ih[(cnn + nlo) * D_ + k1] : 0.f;
    a_ir = __builtin_amdgcn_wmma_f32_16x16x4_f32(false, a, false, br, (short)0, a_ir, false, false);
    a_iz = __builtin_amdgcn_wmma_f32_16x16x4_f32(false, a, false, bz, (short)0, a_iz, false, false);
    a_in = __builtin_amdgcn_wmma_f32_16x16x4_f32(false, a, false, bn, (short)0, a_in, false, false);
  }
  // gh = hn @ W_hh^T   (K = 64)
#pragma unroll
  for (int kb = 0; kb < HID_; kb += 4) {
    int k0 = kb + khalf, k1 = k0 + 1;
    v2f a; a.x = lds_h[nlo][k0]; a.y = lds_h[nlo][k1];
    v2f br, bz, bn;
    br.x = W_hh[(cr  + nlo) * HID_ + k0];  br.y = W_hh[(cr  + nlo) * HID_ + k1];
    bz.x = W_hh[(cz  + nlo) * HID_ + k0];  bz.y = W_hh[(cz  + nlo) * HID_ + k1];
    bn.x = W_hh[(cnn + nlo) * HID_ + k0];  bn.y = W_hh[(cnn + nlo) * HID_ + k1];
    a_hr = __builtin_amdgcn_wmma_f32_16x16x4_f32(false, a, false, br, (short)0, a_hr, false, false);
    a_hz = __builtin_amdgcn_wmma_f32_16x16x4_f32(false, a, false, bz, (short)0, a_hz, false, false);
    a_hn = __builtin_amdgcn_wmma_f32_16x16x4_f32(false, a, false, bn, (short)0, a_hn, false, false);
  }

  float bir = b_ih[cr + nlo], biz = b_ih[cz + nlo], bin_ = b_ih[cnn + nlo];
  float bhr = b_hh[cr + nlo], bhz = b_hh[cz + nlo], bhn = b_hh[cnn + nlo];

#pragma unroll
  for (int v = 0; v < 8; ++v) {
    int Mr = v + ((lane >> 4) << 3);  // C/D layout: M = v (+8 for upper half)
    float ir = a_ir[v] + bir, iz = a_iz[v] + biz, in_ = a_in[v] + bin_;
    float hr = a_hr[v] + bhr, hz = a_hz[v] + bhz, hv = a_hn[v] + bhn;
    float r = 1.f / (1.f + expf(-(ir + hr)));
    float z = 1.f / (1.f + expf(-(iz + hz)));
    float nn = tanhf(in_ + r * hv);
    float hprev = lds_h[Mr][g * 16 + nlo];
    float hnew = (1.f - z) * nn + z * hprev;
    hn[(rowbase + Mr) * HID_ + g * 16 + nlo] = hnew;
    lds_o[Mr][g * 16 + nlo] = hnew;
  }
  __syncthreads();
  if (tid < 16) {
    float s = b_node[0];
#pragma unroll
    for (int c = 0; c < HID_; ++c) s += lds_o[tid][c] * W_node[c];
    node_out[rowbase + tid] = s;
  }
}

// ---------------------------------------------------------------------------
// Per step: per-(b,row) sparse scatter -> softmax -> blend R0 -> softmax
//           -> write R row to d_out and fuse R @ node_out.
// Block = 256 threads = 8 waves; one wave per row.
// ---------------------------------------------------------------------------
__global__ __launch_bounds__(256) void row_kernel(const float* __restrict__ R0,
                                                  const float* __restrict__ a_param,
                                                  const float* __restrict__ e_rep,
                                                  const int* __restrict__ row_ptr,
                                                  const int* __restrict__ csr_col,
                                                  const int* __restrict__ csr_eid,
                                                  const float* __restrict__ node_out,
                                                  float* __restrict__ cn,
                                                  float* __restrict__ out, int t) {
  __shared__ float lds[8][N_];
  int tid = threadIdx.x, lane = tid & 31, w = tid >> 5;
  int g = blockIdx.x * 8 + w;
  int b = g >> 8, n = g & (N_ - 1);
  float alpha = 1.f / (1.f + expf(-a_param[0]));

#pragma unroll
  for (int j = 0; j < 8; ++j) lds[w][j * 32 + lane] = 0.f;
  __syncthreads();
  int s0 = row_ptr[n], s1 = row_ptr[n + 1];
  for (int i = s0 + lane; i < s1; i += 32)
    lds[w][csr_col[i]] = e_rep[b * E_ + csr_eid[i]];
  __syncthreads();

  float v[8], m = -1e30f;
#pragma unroll
  for (int j = 0; j < 8; ++j) { v[j] = lds[w][j * 32 + lane]; m = fmaxf(m, v[j]); }
  for (int o = 16; o; o >>= 1) m = fmaxf(m, __shfl_xor(m, o, 32));
  float s = 0.f;
#pragma unroll
  for (int j = 0; j < 8; ++j) { v[j] = expf(v[j] - m); s += v[j]; }
  for (int o = 16; o; o >>= 1) s += __shfl_xor(s, o, 32);
  float inv = 1.f / s;

  m = -1e30f;
#pragma unroll
  for (int j = 0; j < 8; ++j) {
    v[j] = alpha * R0[n * N_ + j * 32 + lane] + (1.f - alpha) * (v[j] * inv);
    m = fmaxf(m, v[j]);
  }
  for (int o = 16; o; o >>= 1) m = fmaxf(m, __shfl_xor(m, o, 32));
  s = 0.f;
#pragma unroll
  for (int j = 0; j < 8; ++j) { v[j] = expf(v[j] - m); s += v[j]; }
  for (int o = 16; o; o >>= 1) s += __shfl_xor(s, o, 32);
  inv = 1.f / s;

  float* Rout = out + PRED_SZ + (((size_t)(b * PRED_ + t) * N_ + n) * N_);
  float dot = 0.f;
#pragma unroll
  for (int j = 0; j < 8; ++j) {
    float r = v[j] * inv;
    Rout[j * 32 + lane] = r;
    dot += r * node_out[b * N_ + j * 32 + lane];
  }
  for (int o = 16; o; o >>= 1) dot += __shfl_xor(dot, o, 32);
  if (lane == 0) {
    cn[b * N_ + n] = dot;
    out[(b * PRED_ + t) * N_ + n] = dot;
  }
}

// ---------------------------------------------------------------------------
extern "C" void kernel_launch(void* const* d_in, const int* in_sizes, int n_in,
                              void* d_out, int out_size, void* d_ws, size_t ws_size,
                              hipStream_t stream) {
  const float* pm25_hist = (const float*)d_in[0];
  const float* feature   = (const float*)d_in[1];
  const int*   edge_index= (const int*)  d_in[2];
  const float* edge_attr = (const float*)d_in[3];
  const float* wind_mean = (const float*)d_in[4];
  const float* wind_std  = (const float*)d_in[5];
  const float* R0        = (const float*)d_in[6];
  const float* a_param   = (const float*)d_in[7];
  const float* W_ih      = (const float*)d_in[8];
  const float* b_ih      = (const float*)d_in[9];
  const float* W_hh      = (const float*)d_in[10];
  const float* b_hh      = (const float*)d_in[11];
  const float* W_node    = (const float*)d_in[12];
  const float* b_node    = (const float*)d_in[13];
  const float* W_e1      = (const float*)d_in[14];
  const float* b_e1      = (const float*)d_in[15];
  const float* W_e2      = (const float*)d_in[16];
  const float* b_e2      = (const float*)d_in[17];
  float* out = (float*)d_out;

  float* hn       = (float*)d_ws;           // M*HID
  float* cn       = hn + M_ * HID_;         // M
  float* node_out = cn + M_;                // M
  float* e_rep    = node_out + M_;          // B*E
  float* ean      = e_rep + B_ * E_;        // E*2
  int*   row_ptr  = (int*)(ean + E_ * 2);   // N+1
  int*   csr_col  = row_ptr + (N_ + 4);     // E
  int*   csr_eid  = csr_col + E_;           // E

  // zero hn and cn (contiguous)
  hipMemsetAsync(hn, 0, (size_t)(M_ * HID_ + M_) * sizeof(float), stream);

  ean_kernel<<<1, 256, 0, stream>>>(edge_attr, ean);
  csr_kernel<<<1, 256, 0, stream>>>(edge_index, row_ptr, csr_col, csr_eid);

  for (int t = 0; t < PRED_; ++t) {
    edge_kernel<<<(B_ * E_) / 256, 256, 0, stream>>>(
        feature, edge_index, edge_attr, ean, wind_mean, wind_std,
        W_e1, b_e1, W_e2, b_e2, e_rep, t);
    gru_kernel<<<M_ / 16, 128, 0, stream>>>(
        pm25_hist, feature, W_ih, b_ih, W_hh, b_hh, W_node, b_node,
        hn, cn, node_out, t);
    row_kernel<<<(B_ * N_) / 8, 256, 0, stream>>>(
        R0, a_param, e_rep, row_ptr, csr_col, csr_eid, node_out, cn, out, t);
  }
  (void)in_sizes; (void)n_in; (void)out_size; (void)ws_size;
}